// SubPointConv_19430432047362
// MI455X (gfx1250) — compile-verified
//
#include <hip/hip_runtime.h>

typedef float v2f __attribute__((ext_vector_type(2)));
typedef float v8f __attribute__((ext_vector_type(8)));

#define B_    4
#define C_    64
#define N_    4096
#define K_    16
#define HID_  128
#define OUT_  128
#define G_    8
#define TWOC_ 128

// ---------------------------------------------------------------------------
// Kernel 1: fused pairwise-inner-product (fp32 WMMA) + streaming top-16.
// One wave (32 threads) per 16 query points. Never materializes pdist.
// Score used for ranking: 2*<x_q, x_c> - ||x_c||^2 (row term dropped).
// ---------------------------------------------------------------------------
__global__ __launch_bounds__(32)
void knn_topk_kernel(const float* __restrict__ feats, int* __restrict__ knn_idx) {
    __shared__ float sc[16 * 17];           // 16x16 score tile, padded

    const int blk  = blockIdx.x;
    const int b    = blk / (N_ / 16);
    const int q0   = (blk % (N_ / 16)) * 16;
    const int lane = threadIdx.x;           // 0..31 (wave32)
    const int col  = lane & 15;
    const int hi   = lane >> 4;             // half-wave: K pair select
    const int kh   = hi * 2;

    const float* fb = feats + (size_t)b * C_ * N_;

    // A tile: 16 queries x 64 channels, resident in 32 VGPRs.
    v2f a[16];
    {
        const int q = q0 + col;
#pragma unroll
        for (int s = 0; s < 16; ++s) {
            a[s].x = fb[(4 * s + kh) * N_ + q];
            a[s].y = fb[(4 * s + kh + 1) * N_ + q];
        }
    }

    float best_d[16];
    int   best_i[16];
#pragma unroll
    for (int t = 0; t < 16; ++t) { best_d[t] = -3.4e38f; best_i[t] = 0; }

    for (int c0 = 0; c0 < N_; c0 += 16) {
        const int p = c0 + col;
        v2f bb[16];
        float ss = 0.f;                      // ||x_c||^2 partial (this half-wave's channels)
#pragma unroll
        for (int s = 0; s < 16; ++s) {
            bb[s].x = fb[(4 * s + kh) * N_ + p];
            bb[s].y = fb[(4 * s + kh + 1) * N_ + p];
            ss += bb[s].x * bb[s].x + bb[s].y * bb[s].y;
        }
        ss += __shfl_xor(ss, 16, 32);        // full ||x_c||^2 in both half-waves

        v8f acc = {};
#pragma unroll
        for (int s = 0; s < 16; ++s)
            acc = __builtin_amdgcn_wmma_f32_16x16x4_f32(
                false, a[s], false, bb[s], (short)0, acc, false, false);

        // C layout: lane=col(candidate), row = vgpr + 8*hi (query)
#pragma unroll
        for (int v = 0; v < 8; ++v)
            sc[(v + 8 * hi) * 17 + col] = 2.0f * acc[v] - ss;
        __syncthreads();                     // single-wave WG: lowers to s_nop

        if (lane < 16) {                     // lane = query row
#pragma unroll
            for (int j = 0; j < 16; ++j) {
                const float s = sc[lane * 17 + j];
                if (s > best_d[15]) {        // rare path: register insertion sort
                    float d = s; int id = c0 + j;
#pragma unroll
                    for (int t = 0; t < 16; ++t) {
                        const bool  gt = d > best_d[t];
                        const float td = best_d[t];
                        const int   ti = best_i[t];
                        best_d[t] = gt ? d : td;
                        best_i[t] = gt ? id : ti;
                        d  = gt ? td : d;
                        id = gt ? ti : id;
                    }
                }
            }
        }
        __syncthreads();                     // reconverge before next WMMA tile
    }

    if (lane < 16) {
        int* o = knn_idx + ((size_t)b * N_ + q0 + lane) * K_;
#pragma unroll
        for (int t = 0; t < 16; ++t) o[t] = best_i[t];
    }
}

// ---------------------------------------------------------------------------
// Kernel 2: gather graph feature (128x16) -> W1 GEMM + bias + ReLU ->
//           W2 GEMM + bias -> max over 16 neighbors -> (c g) rearrange.
// One block (8 waves) per (batch, point). Each wave owns 16 output rows.
// ---------------------------------------------------------------------------
__global__ __launch_bounds__(256)
void edgeconv_kernel(const float* __restrict__ feats,
                     const int* __restrict__ knn_idx,
                     const float* __restrict__ W1, const float* __restrict__ b1,
                     const float* __restrict__ W2, const float* __restrict__ b2,
                     float* __restrict__ out) {
    __shared__ float Gm[TWOC_ * 17];         // graph matrix [128][16], padded
    __shared__ float Hm[HID_ * 17];          // hidden      [128][16], padded

    const int blk = blockIdx.x;
    const int b   = blk / N_;
    const int n   = blk % N_;
    const int tid = threadIdx.x;

    const float* fb  = feats + (size_t)b * C_ * N_;
    const int*   idx = knn_idx + ((size_t)b * N_ + n) * K_;

    // ---- gather: rows 0..63 = knn - center, rows 64..127 = center ----
    {
        const int j  = tid & 15;
        const int nb = idx[j];
        for (int c = tid >> 4; c < C_; c += 16) {
            const float ctr = fb[c * N_ + n];
            const float nv  = fb[c * N_ + nb];
            Gm[c * 17 + j]        = nv - ctr;
            Gm[(C_ + c) * 17 + j] = ctr;
        }
    }
    __syncthreads();

    const int wave = tid >> 5;               // 0..7 -> 16-row slice of W
    const int lane = tid & 31;
    const int col  = lane & 15;              // neighbor j
    const int hi   = lane >> 4;
    const int kh   = 2 * hi;
    const int m0   = wave * 16;

    // ---- conv1: H = relu(W1 * G + b1) ----
    v8f acc;
#pragma unroll
    for (int v = 0; v < 8; ++v) acc[v] = b1[m0 + v + 8 * hi];

    const float* w1r = W1 + (size_t)(m0 + col) * TWOC_;
#pragma unroll
    for (int s = 0; s < 32; ++s) {
        const v2f av = *(const v2f*)(w1r + 4 * s + kh);   // 8B-aligned
        v2f bv;
        bv.x = Gm[(4 * s + kh) * 17 + col];
        bv.y = Gm[(4 * s + kh + 1) * 17 + col];
        acc = __builtin_amdgcn_wmma_f32_16x16x4_f32(
            false, av, false, bv, (short)0, acc, false, false);
    }
#pragma unroll
    for (int v = 0; v < 8; ++v) {
        const float h = acc[v] > 0.f ? acc[v] : 0.f;
        Hm[(m0 + v + 8 * hi) * 17 + col] = h;
    }
    __syncthreads();

    // ---- conv2: O = W2 * H + b2 ----
#pragma unroll
    for (int v = 0; v < 8; ++v) acc[v] = b2[m0 + v + 8 * hi];

    const float* w2r = W2 + (size_t)(m0 + col) * HID_;
#pragma unroll
    for (int s = 0; s < 32; ++s) {
        const v2f av = *(const v2f*)(w2r + 4 * s + kh);
        v2f bv;
        bv.x = Hm[(4 * s + kh) * 17 + col];
        bv.y = Hm[(4 * s + kh + 1) * 17 + col];
        acc = __builtin_amdgcn_wmma_f32_16x16x4_f32(
            false, av, false, bv, (short)0, acc, false, false);
    }

    // ---- max over 16 neighbors (lane = column) + rearrange store ----
#pragma unroll
    for (int v = 0; v < 8; ++v) {
        float m = acc[v];
#pragma unroll
        for (int off = 8; off >= 1; off >>= 1)
            m = fmaxf(m, __shfl_xor(m, off, 16));
        acc[v] = m;
    }
    if (col == 0) {                          // lanes 0 and 16 write 8 rows each
#pragma unroll
        for (int v = 0; v < 8; ++v) {
            const int row = m0 + v + 8 * hi;
            const int cc  = row >> 3;        // OUT//G channel
            const int g   = row & 7;
            out[(((size_t)b * (OUT_ / G_) + cc) * N_ + n) * G_ + g] = acc[v];
        }
    }
}

// ---------------------------------------------------------------------------
extern "C" void kernel_launch(void* const* d_in, const int* in_sizes, int n_in,
                              void* d_out, int out_size, void* d_ws, size_t ws_size,
                              hipStream_t stream) {
    const float* feats = (const float*)d_in[0];
    const float* W1    = (const float*)d_in[1];
    const float* b1    = (const float*)d_in[2];
    const float* W2    = (const float*)d_in[3];
    const float* b2    = (const float*)d_in[4];
    float* out = (float*)d_out;
    int*   knn = (int*)d_ws;                 // B*N*K ints = 1 MB scratch

    knn_topk_kernel<<<B_ * (N_ / 16), 32, 0, stream>>>(feats, knn);
    edgeconv_kernel<<<B_ * N_, 256, 0, stream>>>(feats, knn, W1, b1, W2, b2, out);
}